// Attention_block_23467701306031
// MI455X (gfx1250) — compile-verified
//
#include <hip/hip_runtime.h>
#include <cstdint>
#include <cstddef>

// MI455X (gfx1250) Attention-gate block.
// Memory-bound (AI ~2 FLOP/B): fp32 throughout; the two 64->32 1x1 convs are
// done with V_WMMA_F32_16X16X4_F32 (A = weight 16x4 tile, B = activation
// 4x16 spatial tile, 16 K-steps over 64 input channels, 2 M-tiles for 32
// output channels). All reductions are fixed-shape LDS trees (deterministic).

typedef __attribute__((ext_vector_type(2))) float v2f;
typedef __attribute__((ext_vector_type(8))) float v8f;

#define NS   131072      // 32*64*64  low-res voxels per channel
#define NSX  1048576     // 64*128*128 hi-res voxels per channel
#define EPSI 1e-5f

__device__ __forceinline__ int imin(int a, int b) { return a < b ? a : b; }
__device__ __forceinline__ float flerp(float lo, float hi, float w) { return lo + w * (hi - lo); }

// ---------------- 1) trilinear downsample x[64][64][128][128] -> xsub[64][32][64][64]
__global__ void k_down(const float* __restrict__ x, float* __restrict__ xsub) {
  int idx = blockIdx.x * 256 + threadIdx.x;          // 64*NS threads exactly
  int v = idx & (NS - 1);
  int c = idx >> 17;
  int w = v & 63, h = (v >> 6) & 63, d = v >> 12;
  const float sd = 63.0f / 31.0f, shw = 127.0f / 63.0f;
  float fd = d * sd, fh = h * shw, fw = w * shw;
  int d0 = imin((int)fd, 63), d1 = imin(d0 + 1, 63); float wd = fd - (float)d0;
  int h0 = imin((int)fh, 127), h1 = imin(h0 + 1, 127); float wh = fh - (float)h0;
  int w0 = imin((int)fw, 127), w1 = imin(w0 + 1, 127); float ww = fw - (float)w0;
  const float* p = x + (size_t)c * NSX;
  const float* p00 = p + (size_t)d0 * 16384 + (size_t)h0 * 128;
  const float* p01 = p + (size_t)d0 * 16384 + (size_t)h1 * 128;
  const float* p10 = p + (size_t)d1 * 16384 + (size_t)h0 * 128;
  const float* p11 = p + (size_t)d1 * 16384 + (size_t)h1 * 128;
  float a = flerp(p00[w0], p00[w1], ww);
  float b = flerp(p01[w0], p01[w1], ww);
  float e = flerp(p10[w0], p10[w1], ww);
  float f = flerp(p11[w0], p11[w1], ww);
  float lo = flerp(a, b, wh);
  float hi = flerp(e, f, wh);
  xsub[(size_t)idx] = flerp(lo, hi, wd);
}

// ---------------- 2) dual 1x1 conv via WMMA: out[o][n] = sum_c W[o][c]*in[c][n] + b[o]
// A 16x4 layout: lane L: row M=L%16, K = kbase + (L>=16?2:0) + vgpr
// B 4x16 layout: lane L: col N=L%16, K = kbase + (L>=16?2:0) + vgpr
// C/D 16x16:     reg r, lanes0-15: M=r, lanes16-31: M=r+8, N=lane%16
__global__ void k_conv(const float* __restrict__ gin, const float* __restrict__ xin,
                       const float* __restrict__ Wg, const float* __restrict__ bg,
                       const float* __restrict__ Wx, const float* __restrict__ bx,
                       float* __restrict__ g1, float* __restrict__ x1) {
  int lane = threadIdx.x & 31;
  int wave = threadIdx.x >> 5;
  int l16  = lane & 15;
  int half = lane >> 4;
  int nbase = (blockIdx.x * 8 + wave) * 16;          // 1024 blocks * 8 waves * 16 cols = NS
  for (int t = 0; t < 2; ++t) {
    const float* in = t ? xin : gin;
    const float* W  = t ? Wx : Wg;
    const float* bb = t ? bx : bg;
    float* outp     = t ? x1 : g1;
    v8f acc0 = {}; v8f acc1 = {};
    for (int k = 0; k < 64; k += 4) {
      int kk = k + half * 2;
      v2f a0, a1, bv;
      a0.x = W[l16 * 64 + kk];        a0.y = W[l16 * 64 + kk + 1];
      a1.x = W[(16 + l16) * 64 + kk]; a1.y = W[(16 + l16) * 64 + kk + 1];
      bv.x = in[(size_t)kk * NS + nbase + l16];
      bv.y = in[(size_t)(kk + 1) * NS + nbase + l16];
      acc0 = __builtin_amdgcn_wmma_f32_16x16x4_f32(false, a0, false, bv, (short)0, acc0, false, false);
      acc1 = __builtin_amdgcn_wmma_f32_16x16x4_f32(false, a1, false, bv, (short)0, acc1, false, false);
    }
    int chl = half * 8;
#pragma unroll
    for (int r = 0; r < 8; ++r) {
      int c0 = r + chl;
      outp[(size_t)c0 * NS + nbase + l16] = acc0[r] + bb[c0];
      int c1 = 16 + c0;
      outp[(size_t)c1 * NS + nbase + l16] = acc1[r] + bb[c1];
    }
  }
}

// ---------------- 3) per-channel partial sums over 64 rows (32 g + 32 x, contiguous)
__global__ void k_stats1(const float* __restrict__ raw, float* __restrict__ part) {
  __shared__ float ss[256], sq[256];
  int row = blockIdx.x >> 3;
  int seg = blockIdx.x & 7;
  const float* p = raw + (size_t)row * NS + (size_t)seg * 16384;
  float s = 0.f, q = 0.f;
  for (int i = threadIdx.x; i < 16384; i += 256) { float v = p[i]; s += v; q += v * v; }
  ss[threadIdx.x] = s; sq[threadIdx.x] = q;
  __syncthreads();
  for (int off = 128; off > 0; off >>= 1) {
    if ((int)threadIdx.x < off) {
      ss[threadIdx.x] += ss[threadIdx.x + off];
      sq[threadIdx.x] += sq[threadIdx.x + off];
    }
    __syncthreads();
  }
  if (threadIdx.x == 0) { part[blockIdx.x * 2] = ss[0]; part[blockIdx.x * 2 + 1] = sq[0]; }
}

// ---------------- 4) finalize per-channel mean / rstd (64 rows)
__global__ void k_fin1(const float* __restrict__ part, float* __restrict__ params) {
  int r = threadIdx.x;                                // 64 threads
  float s = 0.f, q = 0.f;
  for (int seg = 0; seg < 8; ++seg) { s += part[(r * 8 + seg) * 2]; q += part[(r * 8 + seg) * 2 + 1]; }
  float mu  = s * (1.0f / NS);
  float var = fmaxf(q * (1.0f / NS) - mu * mu, 0.f);
  params[r * 2]     = mu;
  params[r * 2 + 1] = rsqrtf(var + EPSI);
}

// ---------------- 5) s = IN(g1)+IN(x1); PReLU; psiraw = W_psi . s + b_psi
__global__ void k_psi(const float* __restrict__ g1, const float* __restrict__ x1,
                      const float* __restrict__ params, const float* __restrict__ Wpsi,
                      const float* __restrict__ bpsi, const float* __restrict__ pa,
                      float* __restrict__ psiraw) {
  __shared__ float sp[128];
  __shared__ float swp[32];
  if (threadIdx.x < 128) sp[threadIdx.x] = params[threadIdx.x];
  if (threadIdx.x < 32)  swp[threadIdx.x] = Wpsi[threadIdx.x];
  __syncthreads();
  int v = blockIdx.x * 256 + threadIdx.x;             // NS threads exactly
  float slope = pa[0];
  float acc = bpsi[0];
#pragma unroll 4
  for (int c = 0; c < 32; ++c) {
    float gg = (g1[(size_t)c * NS + v] - sp[2 * c]) * sp[2 * c + 1];
    float xx = (x1[(size_t)c * NS + v] - sp[64 + 2 * c]) * sp[64 + 2 * c + 1];
    float s = gg + xx;
    s = s >= 0.f ? s : slope * s;
    acc += swp[c] * s;
  }
  psiraw[v] = acc;
}

// ---------------- 6) psi instance-norm stats: single block, deterministic tree
__global__ void k_stats2(const float* __restrict__ pr, float* __restrict__ params2) {
  __shared__ float ss[1024], sq[1024];
  float s = 0.f, q = 0.f;
  for (int i = threadIdx.x; i < NS; i += 1024) { float v = pr[i]; s += v; q += v * v; }
  ss[threadIdx.x] = s; sq[threadIdx.x] = q;
  __syncthreads();
  for (int off = 512; off > 0; off >>= 1) {
    if ((int)threadIdx.x < off) {
      ss[threadIdx.x] += ss[threadIdx.x + off];
      sq[threadIdx.x] += sq[threadIdx.x + off];
    }
    __syncthreads();
  }
  if (threadIdx.x == 0) {
    float mu  = ss[0] * (1.0f / NS);
    float var = fmaxf(sq[0] * (1.0f / NS) - mu * mu, 0.f);
    params2[0] = mu;
    params2[1] = rsqrtf(var + EPSI);
  }
}

// ---------------- 7) gate: xg = xsub * sigmoid((psiraw - mu) * rstd)
__global__ void k_gate(const float* __restrict__ xsub, const float* __restrict__ psiraw,
                       const float* __restrict__ params2, float* __restrict__ xg) {
  int idx = blockIdx.x * 256 + threadIdx.x;            // 64*NS threads exactly
  int v = idx & (NS - 1);
  float z = (psiraw[v] - params2[0]) * params2[1];
  float p = 1.0f / (1.0f + __expf(-z));
  xg[(size_t)idx] = xsub[(size_t)idx] * p;
}

// ---------------- 8) trilinear upsample xg (32,64,64)->(64,128,128) + beta*x
__global__ void k_up(const float* __restrict__ xg, const float* __restrict__ x,
                     const float* __restrict__ beta, float* __restrict__ out) {
  size_t idx = (size_t)blockIdx.x * 256 + threadIdx.x; // 64*NSX threads exactly
  int w = (int)(idx & 127), h = (int)((idx >> 7) & 127), d = (int)((idx >> 14) & 63);
  int c = (int)(idx >> 20);
  const float sd = 31.0f / 63.0f, shw = 63.0f / 127.0f;
  float fd = d * sd, fh = h * shw, fw = w * shw;
  int d0 = imin((int)fd, 31), d1 = imin(d0 + 1, 31); float wd = fd - (float)d0;
  int h0 = imin((int)fh, 63), h1 = imin(h0 + 1, 63); float wh = fh - (float)h0;
  int w0 = imin((int)fw, 63), w1 = imin(w0 + 1, 63); float ww = fw - (float)w0;
  const float* p = xg + (size_t)c * NS;
  const float* p00 = p + (size_t)d0 * 4096 + (size_t)h0 * 64;
  const float* p01 = p + (size_t)d0 * 4096 + (size_t)h1 * 64;
  const float* p10 = p + (size_t)d1 * 4096 + (size_t)h0 * 64;
  const float* p11 = p + (size_t)d1 * 4096 + (size_t)h1 * 64;
  float a = flerp(p00[w0], p00[w1], ww);
  float b = flerp(p01[w0], p01[w1], ww);
  float e = flerp(p10[w0], p10[w1], ww);
  float f = flerp(p11[w0], p11[w1], ww);
  float lo = flerp(a, b, wh);
  float hi = flerp(e, f, wh);
  out[idx] = flerp(lo, hi, wd) + beta[0] * x[idx];
}

extern "C" void kernel_launch(void* const* d_in, const int* in_sizes, int n_in,
                              void* d_out, int out_size, void* d_ws, size_t ws_size,
                              hipStream_t stream) {
  (void)in_sizes; (void)n_in; (void)out_size; (void)ws_size;
  const float* g     = (const float*)d_in[0];   // [1,64,32,64,64]
  const float* x     = (const float*)d_in[1];   // [1,64,64,128,128]
  const float* W_g   = (const float*)d_in[2];   // [32,64]
  const float* b_g   = (const float*)d_in[3];   // [32]
  const float* W_x   = (const float*)d_in[4];   // [32,64]
  const float* b_x   = (const float*)d_in[5];   // [32]
  const float* W_psi = (const float*)d_in[6];   // [1,32]
  const float* b_psi = (const float*)d_in[7];   // [1]
  const float* pa    = (const float*)d_in[8];   // [1]
  const float* beta  = (const float*)d_in[9];   // [1]
  float* out = (float*)d_out;

  // workspace layout (~68 MB); xg aliases g1raw/x1raw (dead after k_psi; fully
  // rewritten by k_gate each call -> replay-deterministic)
  char* ws = (char*)d_ws;
  float* xsub    = (float*)(ws);                                        // 32 MB
  float* g1raw   = (float*)(ws + ((size_t)32 << 20));                   // 16 MB
  float* x1raw   = (float*)(ws + ((size_t)48 << 20));                   // 16 MB
  float* xg      = g1raw;                                               // alias 32 MB
  float* psiraw  = (float*)(ws + ((size_t)64 << 20));                   // 512 KB
  float* part1   = (float*)(ws + ((size_t)64 << 20) + (1 << 19));       // 4 KB
  float* params1 = part1 + 1024;                                        // 512 B
  float* params2 = params1 + 128;                                       // 8 B

  k_down  <<<32768, 256, 0, stream>>>(x, xsub);
  k_conv  <<<1024,  256, 0, stream>>>(g, xsub, W_g, b_g, W_x, b_x, g1raw, x1raw);
  k_stats1<<<512,   256, 0, stream>>>(g1raw, part1);     // g1raw/x1raw contiguous -> 64 rows
  k_fin1  <<<1,      64, 0, stream>>>(part1, params1);
  k_psi   <<<512,   256, 0, stream>>>(g1raw, x1raw, params1, W_psi, b_psi, pa, psiraw);
  k_stats2<<<1,    1024, 0, stream>>>(psiraw, params2);
  k_gate  <<<32768, 256, 0, stream>>>(xsub, psiraw, params2, xg);
  k_up    <<<262144,256, 0, stream>>>(xg, x, beta, out);
}